// PDEDiffusionModel_10393820857052
// MI455X (gfx1250) — compile-verified
//
#include <hip/hip_runtime.h>
#include <math.h>
#include <stdint.h>

#define BB 16
#define HH 512
#define WW 512
#define NPLANE (HH*WW)

#define DTC 0.01f
#define NUC 0.001f
#define TWO_PI 6.28318530717958647692f
#define INV_N2 (1.0f / 262144.0f)   // 1/(512*512) irfft2 normalization

typedef float v2f __attribute__((ext_vector_type(2)));
typedef float v8f __attribute__((ext_vector_type(8)));

__device__ __forceinline__ int wrapm(int i, int n) { return (i == 0) ? (n - 1) : (i - 1); }
__device__ __forceinline__ int wrapp(int i, int n) { return (i == n - 1) ? 0 : (i + 1); }

// ---------------------------------------------------------------------------
// Async global -> LDS staging (CDNA5 GLOBAL_LOAD_ASYNC_TO_LDS_B32, ASYNCcnt).
// Guarded: falls back to plain copies if the builtins are not declared.
// The builtin takes (addrspace(1) int*, addrspace(3) int*, imm offset, imm cpol).
// ---------------------------------------------------------------------------
#if __has_builtin(__builtin_amdgcn_global_load_async_to_lds_b32) && \
    __has_builtin(__builtin_amdgcn_s_wait_asynccnt)
#define ASYNC_LDS 1
#else
#define ASYNC_LDS 0
#endif

#if ASYNC_LDS
__device__ __forceinline__ __attribute__((address_space(3))) int* to_lds(void* p)
{
    // flat->LDS addrspace lowering on AMDGPU is a low-32-bit truncation
    return (__attribute__((address_space(3))) int*)(uint32_t)(uintptr_t)p;
}
__device__ __forceinline__ __attribute__((address_space(1))) int* to_gbl(const void* p)
{
    return (__attribute__((address_space(1))) int*)(uintptr_t)p;
}
#endif

__device__ __forceinline__ void async_ld32(float* ldsDst, const float* g)
{
#if ASYNC_LDS
    __builtin_amdgcn_global_load_async_to_lds_b32(to_gbl(g), to_lds(ldsDst), 0, 0);
#else
    *ldsDst = *g;
#endif
}

__device__ __forceinline__ void wait_async()
{
#if ASYNC_LDS
    __builtin_amdgcn_s_wait_asynccnt(0);
#endif
}

// ---------------------------------------------------------------------------
// Per-wave 512-point complex FFT in LDS.
//   512 = 16 x 32.  Stage 1: radix-16 via WMMA f32 16x16x4 (complex matmul
//   D16(16x16) * X(16x32)), twiddle fused into writeback.  Stage 2: 16
//   independent 32-point radix-2 DIT transforms.
//   dsign = +1 forward (e^{-i...}), -1 inverse.
//   Output element X[o] lives at lds[(o&15)*32 + (o>>4)].
// ---------------------------------------------------------------------------
__device__ __forceinline__ void wave_fft512(float* re, float* im, int lane, float dsign)
{
    const int mrow = lane & 15;   // A-matrix M row for this lane
    const int kh   = lane >> 4;   // lane-half selects K sub-rows

    // DFT-16 matrix A-operand slices (16x4 each), per documented layout.
    // An = -Ai (f32 WMMA NEG field is C-only; no A negate).
    v2f Ar[4], Ai[4], An[4];
#pragma unroll
    for (int s = 0; s < 4; ++s) {
#pragma unroll
        for (int j = 0; j < 2; ++j) {
            int K = 4 * s + 2 * kh + j;
            float ang = -TWO_PI * (float)(mrow * K) * (1.0f / 16.0f) * dsign;
            float sv, cv;
            __sincosf(ang, &sv, &cv);
            Ar[s][j] = cv;
            Ai[s][j] = sv;
            An[s][j] = -sv;
        }
    }

    v8f Cr[2], Ci[2];
    Cr[0] = (v8f){}; Cr[1] = (v8f){}; Ci[0] = (v8f){}; Ci[1] = (v8f){};

#pragma unroll
    for (int blk = 0; blk < 2; ++blk) {
        const int n = 16 * blk + (lane & 15);
#pragma unroll
        for (int s = 0; s < 4; ++s) {
            v2f Br, Bi;
#pragma unroll
            for (int j = 0; j < 2; ++j) {
                int K = 4 * s + 2 * kh + j;
                Br[j] = re[K * 32 + n];
                Bi[j] = im[K * 32 + n];
            }
            // Yr += Dr*Xr + (-Di)*Xi ; Yi += Dr*Xi + Di*Xr
            Cr[blk] = __builtin_amdgcn_wmma_f32_16x16x4_f32(false, Ar[s], false, Br, (short)0, Cr[blk], false, false);
            Cr[blk] = __builtin_amdgcn_wmma_f32_16x16x4_f32(false, An[s], false, Bi, (short)0, Cr[blk], false, false);
            Ci[blk] = __builtin_amdgcn_wmma_f32_16x16x4_f32(false, Ar[s], false, Bi, (short)0, Ci[blk], false, false);
            Ci[blk] = __builtin_amdgcn_wmma_f32_16x16x4_f32(false, Ai[s], false, Br, (short)0, Ci[blk], false, false);
        }
    }

    // Writeback with inter-stage twiddle W512^{k1*n2}.
#pragma unroll
    for (int blk = 0; blk < 2; ++blk) {
        const int n = 16 * blk + (lane & 15);
#pragma unroll
        for (int r = 0; r < 8; ++r) {
            int mm = r + 8 * kh;
            float ang = -TWO_PI * (float)(mm * n) * (1.0f / 512.0f) * dsign;
            float sv, cv;
            __sincosf(ang, &sv, &cv);
            float yr = Cr[blk][r], yi = Ci[blk][r];
            re[mm * 32 + n] = yr * cv - yi * sv;
            im[mm * 32 + n] = yr * sv + yi * cv;
        }
    }

    // Stage 2: 16 rows x 32-point DFT (radix-2 DIT, bit-reversed input).
    float tr[16], ti[16];
#pragma unroll
    for (int i = 0; i < 16; ++i) {
        int g = lane + 32 * i;
        int row = g >> 5, col = g & 31;
        int rc = (int)(__brev((unsigned)col) >> 27);
        tr[i] = re[row * 32 + rc];
        ti[i] = im[row * 32 + rc];
    }
#pragma unroll
    for (int i = 0; i < 16; ++i) {
        int g = lane + 32 * i;
        re[g] = tr[i];
        im[g] = ti[i];
    }
#pragma unroll
    for (int st = 1; st <= 5; ++st) {
        int mspan = 1 << st, half = mspan >> 1;
#pragma unroll
        for (int i = 0; i < 8; ++i) {
            int bf = lane + 32 * i;
            int row = bf >> 4;
            int q = bf & 15;
            int kg = q >> (st - 1);
            int j = q & (half - 1);
            int a = row * 32 + kg * mspan + j;
            int b2 = a + half;
            float ang = -TWO_PI * (float)j / (float)mspan * dsign;
            float sv, cv;
            __sincosf(ang, &sv, &cv);
            float xr = re[b2], xi = im[b2];
            float wr = xr * cv - xi * sv;
            float wi = xr * sv + xi * cv;
            float ar = re[a], ai = im[a];
            re[b2] = ar - wr; im[b2] = ai - wi;
            re[a]  = ar + wr; im[a]  = ai + wi;
        }
    }
}

// ---------------------------------------------------------------------------
// FFT kernels. Complex plane layout per batch: re[512*512] then im[512*512].
// ---------------------------------------------------------------------------
#define LDS_WAVE_STRIDE 1026   // 513 re + 513 im (pad keeps column kernel conflict-free)

__global__ void fft_rows_r2c(const float* __restrict__ src, const float* __restrict__ scaleB,
                             float* __restrict__ dstC)
{
    extern __shared__ float lds[];
    int wave = threadIdx.x >> 5, lane = threadIdx.x & 31;
    int row = blockIdx.x * 8 + wave;
    int b = row >> 9, y = row & 511;
    float* re = lds + wave * LDS_WAVE_STRIDE;
    float* im = re + 513;
    float s = scaleB ? scaleB[b] : 1.0f;
    const float* p = src + (size_t)b * NPLANE + (size_t)y * WW;
#pragma unroll
    for (int i = 0; i < 16; ++i) { int x = lane + 32 * i; async_ld32(re + x, p + x); }
    wait_async();
#pragma unroll
    for (int i = 0; i < 16; ++i) { int x = lane + 32 * i; re[x] *= s; im[x] = 0.0f; }
    wave_fft512(re, im, lane, 1.0f);
    float* dre = dstC + (size_t)b * 2 * NPLANE + (size_t)y * WW;
    float* dim = dre + NPLANE;
#pragma unroll
    for (int i = 0; i < 16; ++i) {
        int o = lane + 32 * i;
        int a = (o & 15) * 32 + (o >> 4);
        dre[o] = re[a];
        dim[o] = im[a];
    }
}

__global__ void fft_rows_c2r(const float* __restrict__ srcC, float* __restrict__ dst)
{
    extern __shared__ float lds[];
    int wave = threadIdx.x >> 5, lane = threadIdx.x & 31;
    int row = blockIdx.x * 8 + wave;
    int b = row >> 9, y = row & 511;
    float* re = lds + wave * LDS_WAVE_STRIDE;
    float* im = re + 513;
    const float* sre = srcC + (size_t)b * 2 * NPLANE + (size_t)y * WW;
    const float* sim = sre + NPLANE;
#pragma unroll
    for (int i = 0; i < 16; ++i) {
        int x = lane + 32 * i;
        async_ld32(re + x, sre + x);
        async_ld32(im + x, sim + x);
    }
    wait_async();
    wave_fft512(re, im, lane, -1.0f);
    float* d = dst + (size_t)b * NPLANE + (size_t)y * WW;
#pragma unroll
    for (int i = 0; i < 16; ++i) {
        int o = lane + 32 * i;
        int a = (o & 15) * 32 + (o >> 4);
        d[o] = re[a];   // 1/N^2 folded into the spectral multiplier
    }
}

// In-place column FFT over a 512x32 tile, 32 waves/WG, coalesced tile I/O.
__global__ void fft_cols_c2c(float* __restrict__ C, float dsign)
{
    extern __shared__ float lds[];
    int lane = threadIdx.x & 31;
    int wv = threadIdx.x >> 5;
    int b = blockIdx.x >> 4;
    int c0 = (blockIdx.x & 15) * 32;
    float* bre = C + (size_t)b * 2 * NPLANE;
    float* bim = bre + NPLANE;
    // Cooperative async load: lanes sweep columns (coalesced); LDS stride 1026
    // words -> bank = (2*col + row) % 64, conflict-free across 32 columns.
#pragma unroll
    for (int i = 0; i < 16; ++i) {
        int row = wv + 32 * i;
        size_t g = (size_t)row * WW + c0 + lane;
        async_ld32(&lds[lane * LDS_WAVE_STRIDE + row], bre + g);
        async_ld32(&lds[lane * LDS_WAVE_STRIDE + 513 + row], bim + g);
    }
    wait_async();        // each wave drains its own ASYNCcnt before the barrier
    __syncthreads();
    {
        float* re = lds + wv * LDS_WAVE_STRIDE;
        float* im = re + 513;
        wave_fft512(re, im, lane, dsign);
    }
    __syncthreads();
#pragma unroll
    for (int i = 0; i < 16; ++i) {
        int o = wv + 32 * i;
        int a = (o & 15) * 32 + (o >> 4);
        size_t g = (size_t)o * WW + c0 + lane;
        bre[g] = lds[lane * LDS_WAVE_STRIDE + a];
        bim[g] = lds[lane * LDS_WAVE_STRIDE + 513 + a];
    }
}

// ---------------------------------------------------------------------------
// Spectral multipliers (full-FFT equivalents of the rfft2 reference; kx uses
// the rfftfreq convention (+256 at Nyquist), ky uses fftfreq (-256)).
// The inverse-transform 1/N^2 normalization is folded in here.
// ---------------------------------------------------------------------------
__global__ void poisson_spectral(float* __restrict__ C)
{
    size_t i = (size_t)blockIdx.x * 256 + threadIdx.x;
    int b = (int)(i / NPLANE);
    int r = (int)(i % NPLANE);
    int y = r >> 9, x = r & 511;
    int kx = (x <= 256) ? x : x - 512;
    int ky = (y < 256) ? y : y - 512;
    float k2 = fmaxf((float)(kx * kx + ky * ky), 1e-10f);
    float m = INV_N2 / (-39.47841760435743f * k2);   // 1/(-4*pi^2*k2) * 1/N^2
    if (x == 0 && y == 0) m = 0.0f;
    float* re = C + (size_t)b * 2 * NPLANE + r;
    float* im = re + NPLANE;
    *re *= m;
    *im *= m;
}

__global__ void project_spectral(float* __restrict__ CX, float* __restrict__ CY)
{
    size_t i = (size_t)blockIdx.x * 256 + threadIdx.x;
    int b = (int)(i / NPLANE);
    int r = (int)(i % NPLANE);
    int y = r >> 9, x = r & 511;
    float kx = (float)((x <= 256) ? x : x - 512);
    float ky = (float)((y < 256) ? y : y - 512);
    float k2 = fmaxf(kx * kx + ky * ky, 1e-10f);
    float inv = 1.0f / k2;
    float* jxr = CX + (size_t)b * 2 * NPLANE + r; float* jxi = jxr + NPLANE;
    float* jyr = CY + (size_t)b * 2 * NPLANE + r; float* jyi = jyr + NPLANE;
    float dr = kx * *jxr + ky * *jyr;
    float di = kx * *jxi + ky * *jyi;
    *jxr = (*jxr - kx * dr * inv) * INV_N2;
    *jxi = (*jxi - kx * di * inv) * INV_N2;
    *jyr = (*jyr - ky * dr * inv) * INV_N2;
    *jyi = (*jyi - ky * di * inv) * INV_N2;
}

// ---------------------------------------------------------------------------
// Scalars, xt, smoothing, reductions, stencils.
// ws scalar block: [0]=s_alpha[16] [16]=s_one [32]=sqrt_beta [48]=cu [64]=cv
//                  [80]=normsq_u accum [96]=normsq_v accum
// ---------------------------------------------------------------------------
__global__ void scalars_kernel(const int* __restrict__ t, float* __restrict__ sc)
{
    int b = threadIdx.x;
    if (b >= BB) return;
    const float S = 0.008f;
    int tb = t[b];
    float fprev = cosf((S / (1.0f + S)) * 1.57079632679489662f);
    fprev *= fprev;
    float acp = 1.0f, beta_t = 0.0f;
    for (int i = 0; i < 1000; ++i) {
        float a = (((float)(i + 1) / 1000.0f + S) / (1.0f + S)) * 1.57079632679489662f;
        float fi = cosf(a); fi *= fi;
        float beta = 1.0f - fi / fprev;
        beta = fminf(fmaxf(beta, 0.0f), 0.999f);
        fprev = fi;
        if (i <= tb) acp *= (1.0f - beta);
        if (i == tb) beta_t = beta;
    }
    sc[b] = sqrtf(acp);
    sc[16 + b] = sqrtf(1.0f - acp);
    sc[32 + b] = sqrtf(beta_t);
}

__global__ void zero_acc_kernel(float* __restrict__ sc)
{
    int i = threadIdx.x;
    if (i < 32) sc[80 + i] = 0.0f;
}

// xt for channels 0..2 only (u,v,p); field channels are recomputed inline.
__global__ void xt_kernel(const float* __restrict__ x0, const float* __restrict__ noise,
                          const float* __restrict__ sc, float* __restrict__ xt)
{
    size_t i = (size_t)blockIdx.x * 256 + threadIdx.x;  // over B*3*NPLANE
    int b = (int)(i / ((size_t)3 * NPLANE));
    int c = (int)((i / NPLANE) % 3);
    size_t r = i % NPLANE;
    size_t src = ((size_t)b * 6 + c) * NPLANE + r;
    xt[i] = sc[b] * x0[src] + sc[16 + b] * noise[src];
}

__global__ void smooth_kernel(const float* __restrict__ src, float* __restrict__ dst)
{
    int x = blockIdx.x * 32 + (threadIdx.x & 31);
    int y = blockIdx.y * 8 + (threadIdx.x >> 5);
    int b = blockIdx.z;
    const float* p = src + (size_t)b * NPLANE;
    int xm = wrapm(x, WW), xp = wrapp(x, WW), ym = wrapm(y, HH), yp = wrapp(y, HH);
    float c = p[y * WW + x];
    float lap = p[y * WW + xm] + p[y * WW + xp] + p[ym * WW + x] + p[yp * WW + x] - 4.0f * c;
    dst[(size_t)b * NPLANE + y * WW + x] = c + 0.1f * lap;
}

__global__ void sumsq_kernel(const float* __restrict__ f, float* __restrict__ acc)
{
    int b = blockIdx.y;
    const float* p = f + (size_t)b * NPLANE;
    float s = 0.0f;
    for (int i = blockIdx.x * 256 + threadIdx.x; i < NPLANE; i += gridDim.x * 256) s += p[i] * p[i];
    __shared__ float red[256];
    red[threadIdx.x] = s;
    __syncthreads();
    for (int w = 128; w > 0; w >>= 1) {
        if (threadIdx.x < w) red[threadIdx.x] += red[threadIdx.x + w];
        __syncthreads();
    }
    if (threadIdx.x == 0) atomicAdd(&acc[b], red[0]);
}

__global__ void coef_kernel(float* __restrict__ sc)
{
    int b = threadIdx.x;
    if (b >= BB) return;
    sc[48 + b] = sc[32 + b] / fmaxf(sqrtf(sc[80 + b]), 1e-12f);
    sc[64 + b] = sc[32 + b] / fmaxf(sqrtf(sc[96 + b]), 1e-12f);
}

__global__ void ustar_kernel(const float* __restrict__ xt, const float* __restrict__ fu,
                             const float* __restrict__ fv, const float* __restrict__ sc,
                             float* __restrict__ us, float* __restrict__ vs)
{
    int x = blockIdx.x * 32 + (threadIdx.x & 31);
    int y = blockIdx.y * 8 + (threadIdx.x >> 5);
    int b = blockIdx.z;
    const float* U = xt + ((size_t)b * 3 + 0) * NPLANE;
    const float* V = xt + ((size_t)b * 3 + 1) * NPLANE;
    const float* P = xt + ((size_t)b * 3 + 2) * NPLANE;
    int xm = wrapm(x, WW), xp = wrapp(x, WW), ym = wrapm(y, HH), yp = wrapp(y, HH);
    int i = y * WW + x;
    float u = U[i], v = V[i];
    float ddxu = 0.5f * (U[y * WW + xp] - U[y * WW + xm]);
    float ddyu = 0.5f * (U[yp * WW + x] - U[ym * WW + x]);
    float lapu = U[y * WW + xm] + U[y * WW + xp] + U[ym * WW + x] + U[yp * WW + x] - 4.0f * u;
    float ddxv = 0.5f * (V[y * WW + xp] - V[y * WW + xm]);
    float ddyv = 0.5f * (V[yp * WW + x] - V[ym * WW + x]);
    float lapv = V[y * WW + xm] + V[y * WW + xp] + V[ym * WW + x] + V[yp * WW + x] - 4.0f * v;
    float dpdx = 0.5f * (P[y * WW + xp] - P[y * WW + xm]);
    float dpdy = 0.5f * (P[yp * WW + x] - P[ym * WW + x]);
    size_t o = (size_t)b * NPLANE + i;
    us[o] = u + DTC * (-(u * ddxu + v * ddyu) + NUC * lapu - dpdx) + sc[48 + b] * fu[o];
    vs[o] = v + DTC * (-(u * ddxv + v * ddyv) + NUC * lapv - dpdy) + sc[64 + b] * fv[o];
}

__global__ void div_kernel(const float* __restrict__ us, const float* __restrict__ vs,
                           float* __restrict__ dv)
{
    int x = blockIdx.x * 32 + (threadIdx.x & 31);
    int y = blockIdx.y * 8 + (threadIdx.x >> 5);
    int b = blockIdx.z;
    const float* U = us + (size_t)b * NPLANE;
    const float* V = vs + (size_t)b * NPLANE;
    int xm = wrapm(x, WW), xp = wrapp(x, WW), ym = wrapm(y, HH), yp = wrapp(y, HH);
    dv[(size_t)b * NPLANE + y * WW + x] =
        0.5f * (U[y * WW + xp] - U[y * WW + xm]) + 0.5f * (V[yp * WW + x] - V[ym * WW + x]);
}

__global__ void finalize_uvp_kernel(const float* __restrict__ us, const float* __restrict__ vs,
                                    const float* __restrict__ xt, const float* __restrict__ pc,
                                    float* __restrict__ out)
{
    int x = blockIdx.x * 32 + (threadIdx.x & 31);
    int y = blockIdx.y * 8 + (threadIdx.x >> 5);
    int b = blockIdx.z;
    const float* PC = pc + (size_t)b * NPLANE;
    int xm = wrapm(x, WW), xp = wrapp(x, WW), ym = wrapm(y, HH), yp = wrapp(y, HH);
    int i = y * WW + x;
    size_t o = (size_t)b * NPLANE + i;
    float ddxp = 0.5f * (PC[y * WW + xp] - PC[y * WW + xm]);
    float ddyp = 0.5f * (PC[yp * WW + x] - PC[ym * WW + x]);
    const float* P = xt + ((size_t)b * 3 + 2) * NPLANE;
    out[((size_t)b * 6 + 0) * NPLANE + i] = us[o] - ddxp;
    out[((size_t)b * 6 + 1) * NPLANE + i] = vs[o] - ddyp;
    out[((size_t)b * 6 + 2) * NPLANE + i] = P[i] + PC[i];
}

__global__ void bz_kernel(const float* __restrict__ x0, const float* __restrict__ noise,
                          const float* __restrict__ sc, float* __restrict__ out)
{
    int x = blockIdx.x * 32 + (threadIdx.x & 31);
    int y = blockIdx.y * 8 + (threadIdx.x >> 5);
    int b = blockIdx.z;
    float sa = sc[b], so = sc[16 + b];
    size_t b3 = ((size_t)b * 6 + 3) * NPLANE;
    size_t b4 = ((size_t)b * 6 + 4) * NPLANE;
    size_t b5 = ((size_t)b * 6 + 5) * NPLANE;
    int xm = wrapm(x, WW), xp = wrapp(x, WW), ym = wrapm(y, HH), yp = wrapp(y, HH);
    auto EX = [&](int yy, int xx) { size_t k = b3 + (size_t)yy * WW + xx; return sa * x0[k] + so * noise[k]; };
    auto EY = [&](int yy, int xx) { size_t k = b4 + (size_t)yy * WW + xx; return sa * x0[k] + so * noise[k]; };
    size_t k5 = b5 + (size_t)y * WW + x;
    float Bz = sa * x0[k5] + so * noise[k5];
    float curl = 0.5f * (EY(y, xp) - EY(y, xm)) - 0.5f * (EX(yp, x) - EX(ym, x));
    out[((size_t)b * 6 + 5) * NPLANE + y * WW + x] = Bz - DTC * curl;   // DT/MU, MU=1
}

__global__ void efield_kernel(const float* __restrict__ x0, const float* __restrict__ noise,
                              const float* __restrict__ sc, const float* __restrict__ jxp,
                              const float* __restrict__ jyp, float* __restrict__ out)
{
    int x = blockIdx.x * 32 + (threadIdx.x & 31);
    int y = blockIdx.y * 8 + (threadIdx.x >> 5);
    int b = blockIdx.z;
    float sa = sc[b], so = sc[16 + b];
    size_t b3 = ((size_t)b * 6 + 3) * NPLANE;
    size_t b4 = ((size_t)b * 6 + 4) * NPLANE;
    const float* BZ = out + ((size_t)b * 6 + 5) * NPLANE;
    int xm = wrapm(x, WW), xp = wrapp(x, WW), ym = wrapm(y, HH), yp = wrapp(y, HH);
    int i = y * WW + x;
    size_t o = (size_t)b * NPLANE + i;
    float Ex = sa * x0[b3 + i] + so * noise[b3 + i];
    float Ey = sa * x0[b4 + i] + so * noise[b4 + i];
    float ddyBz = 0.5f * (BZ[yp * WW + x] - BZ[ym * WW + x]);
    float ddxBz = 0.5f * (BZ[y * WW + xp] - BZ[y * WW + xm]);
    out[((size_t)b * 6 + 3) * NPLANE + i] = Ex + DTC * ddyBz - DTC * jxp[o];  // DT/EPS, EPS=1
    out[((size_t)b * 6 + 4) * NPLANE + i] = Ey - DTC * ddxBz - DTC * jyp[o];
}

// ---------------------------------------------------------------------------
extern "C" void kernel_launch(void* const* d_in, const int* in_sizes, int n_in,
                              void* d_out, int out_size, void* d_ws, size_t ws_size,
                              hipStream_t stream)
{
    (void)in_sizes; (void)n_in; (void)out_size; (void)ws_size;
    const float* x0    = (const float*)d_in[0];
    const float* noise = (const float*)d_in[1];
    const float* f_u   = (const float*)d_in[2];
    const float* f_v   = (const float*)d_in[3];
    const float* jx    = (const float*)d_in[4];
    const float* jy    = (const float*)d_in[5];
    const int*   t     = (const int*)d_in[6];
    float* out = (float*)d_out;

    const size_t BP = (size_t)BB * NPLANE;
    float* ws  = (float*)d_ws;
    float* SC  = ws;                 // 256 floats of scalars
    float* XT  = SC + 256;           // [B][3][N]
    float* FUA = XT + 3 * BP;
    float* FUB = FUA + BP;
    float* FVA = FUB + BP;
    float* FVB = FVA + BP;
    float* UST = FVB + BP;
    float* VST = UST + BP;
    float* R1  = VST + BP;           // div, then p_corr
    float* JXP = R1 + BP;
    float* JYP = JXP + BP;
    float* CA  = JYP + BP;           // complex [B][2][N]
    float* CB  = CA + 2 * BP;

    const int rowsShm = 8 * LDS_WAVE_STRIDE * sizeof(float);
    const int colsShm = 32 * LDS_WAVE_STRIDE * sizeof(float);
    (void)hipFuncSetAttribute((const void*)fft_rows_r2c,  hipFuncAttributeMaxDynamicSharedMemorySize, rowsShm);
    (void)hipFuncSetAttribute((const void*)fft_rows_c2r,  hipFuncAttributeMaxDynamicSharedMemorySize, rowsShm);
    (void)hipFuncSetAttribute((const void*)fft_cols_c2c,  hipFuncAttributeMaxDynamicSharedMemorySize, colsShm);

    dim3 blk(256);
    dim3 g2d(WW / 32, HH / 8, BB);
    dim3 gRows(BB * HH / 8);
    dim3 gCols(BB * (WW / 32));
    int gElem3 = (int)((3 * BP) / 256);
    int gElemC = (int)(BP / 256);

    // 1) diffusion scalars + accum init
    scalars_kernel<<<1, 32, 0, stream>>>(t, SC);
    zero_acc_kernel<<<1, 32, 0, stream>>>(SC);

    // 2) xt for u,v,p
    xt_kernel<<<gElem3, blk, 0, stream>>>(x0, noise, SC, XT);

    // 3) force smoothing (3 Jacobi steps each)
    smooth_kernel<<<g2d, blk, 0, stream>>>(f_u, FUA);
    smooth_kernel<<<g2d, blk, 0, stream>>>(FUA, FUB);
    smooth_kernel<<<g2d, blk, 0, stream>>>(FUB, FUA);
    smooth_kernel<<<g2d, blk, 0, stream>>>(f_v, FVA);
    smooth_kernel<<<g2d, blk, 0, stream>>>(FVA, FVB);
    smooth_kernel<<<g2d, blk, 0, stream>>>(FVB, FVA);

    // 4) L2 norms -> combined coefficients sqrt_beta / ||f||
    sumsq_kernel<<<dim3(64, BB), blk, 0, stream>>>(FUA, SC + 80);
    sumsq_kernel<<<dim3(64, BB), blk, 0, stream>>>(FVA, SC + 96);
    coef_kernel<<<1, 32, 0, stream>>>(SC);

    // 5) provisional velocity + divergence
    ustar_kernel<<<g2d, blk, 0, stream>>>(XT, FUA, FVA, SC, UST, VST);
    div_kernel<<<g2d, blk, 0, stream>>>(UST, VST, R1);

    // 6) Poisson solve via WMMA FFT
    fft_rows_r2c<<<gRows, blk, rowsShm, stream>>>(R1, nullptr, CA);
    fft_cols_c2c<<<gCols, dim3(1024), colsShm, stream>>>(CA, 1.0f);
    poisson_spectral<<<(int)(BP / 256), blk, 0, stream>>>(CA);
    fft_cols_c2c<<<gCols, dim3(1024), colsShm, stream>>>(CA, -1.0f);
    fft_rows_c2r<<<gRows, blk, rowsShm, stream>>>(CA, R1);   // R1 = p_corr

    // 7) final u,v,p
    finalize_uvp_kernel<<<g2d, blk, 0, stream>>>(UST, VST, XT, R1, out);

    // 8) current projection (J scaled by sqrt_beta in the forward row pass)
    fft_rows_r2c<<<gRows, blk, rowsShm, stream>>>(jx, SC + 32, CA);
    fft_rows_r2c<<<gRows, blk, rowsShm, stream>>>(jy, SC + 32, CB);
    fft_cols_c2c<<<gCols, dim3(1024), colsShm, stream>>>(CA, 1.0f);
    fft_cols_c2c<<<gCols, dim3(1024), colsShm, stream>>>(CB, 1.0f);
    project_spectral<<<gElemC, blk, 0, stream>>>(CA, CB);
    fft_cols_c2c<<<gCols, dim3(1024), colsShm, stream>>>(CA, -1.0f);
    fft_cols_c2c<<<gCols, dim3(1024), colsShm, stream>>>(CB, -1.0f);
    fft_rows_c2r<<<gRows, blk, rowsShm, stream>>>(CA, JXP);
    fft_rows_c2r<<<gRows, blk, rowsShm, stream>>>(CB, JYP);

    // 9) Maxwell update
    bz_kernel<<<g2d, blk, 0, stream>>>(x0, noise, SC, out);
    efield_kernel<<<g2d, blk, 0, stream>>>(x0, noise, SC, JXP, JYP, out);
}